// EnhancedLinkPredictionGNN_33346126086480
// MI455X (gfx1250) — compile-verified
//
#include <hip/hip_runtime.h>
#include <hip/hip_bf16.h>

#define HD 32
#define BN_EPS 1e-5f

typedef float v2f __attribute__((ext_vector_type(2)));
typedef float v8f __attribute__((ext_vector_type(8)));

#if __has_builtin(__builtin_amdgcn_wmma_f32_16x16x4_f32)
#define USE_WMMA_F32X4 1
#else
#define USE_WMMA_F32X4 0
#endif

__device__ __forceinline__ void atomAdd(float* p, float v) {
  // non-returning fp32 atomic -> global_atomic_add_f32, executed at L2
  unsafeAtomicAdd(p, v);
}

// ---------------------------------------------------------------- utilities
__global__ void k_zero(float* __restrict__ p, int n) {
  int i = blockIdx.x * blockDim.x + threadIdx.x;
  if (i < n) p[i] = 0.0f;
}

__global__ void k_degree(const int* __restrict__ dst, float* __restrict__ deg, int E) {
  int e = blockIdx.x * blockDim.x + threadIdx.x;
  if (e < E) atomAdd(&deg[dst[e]], 1.0f);
}

__global__ void k_norm(const float* __restrict__ deg, float* __restrict__ nrm, int n) {
  int i = blockIdx.x * blockDim.x + threadIdx.x;
  if (i < n) nrm[i] = rsqrtf(deg[i] + 1.0f);
}

// ------------------------------------------------- layer 1: [N,1]x[1,32] outer product
__global__ void k_l1(const float* __restrict__ x, const float* __restrict__ W1,
                     float* __restrict__ out, int N) {
  long t = (long)blockIdx.x * blockDim.x + threadIdx.x;
  if (t >= (long)N * HD) return;
  out[t] = x[t >> 5] * W1[t & (HD - 1)];
}

// ------------------------------------------------- [N,32]x[32,32] GEMM via V_WMMA_F32_16X16X4_F32
// One wave computes a 16-row x 32-col tile: two 16x16 accumulators, 8 K-steps of 4.
__global__ void k_gemm_wmma(const float* __restrict__ A,   // [N,32]
                            const float* __restrict__ W,   // [32,32] row-major
                            float* __restrict__ Out, int N) {
  const int wave = threadIdx.x >> 5;
  const int lane = threadIdx.x & 31;
  const long row0 = ((long)blockIdx.x * 8 + wave) * 16;
  if (row0 >= N) return;               // wave-uniform: EXEC stays all-ones for WMMA

  const int nc = lane & 15;            // D: N column within 16-tile
  const int mo = (lane >> 4) * 8;      // D: M offset (0 or 8)

#if USE_WMMA_F32X4
  const int m  = lane & 15;            // A/B fragment row/col
  const int kb = (lane >> 4) << 1;     // lanes 16-31 carry K+2,K+3

  // B fragments: W[k][col], col tile c covers columns c*16..c*16+15
  v2f bfrag[2][8];
  #pragma unroll
  for (int c = 0; c < 2; ++c)
    #pragma unroll
    for (int kk = 0; kk < 8; ++kk) {
      const int k = kk * 4 + kb;
      bfrag[c][kk].x = W[k * HD + c * 16 + m];
      bfrag[c][kk].y = W[(k + 1) * HD + c * 16 + m];
    }

  long rl = row0 + m; if (rl > (long)N - 1) rl = (long)N - 1;  // clamp tail loads
  const float* arow = A + rl * HD;

  v8f acc0 = {0.f,0.f,0.f,0.f,0.f,0.f,0.f,0.f};
  v8f acc1 = {0.f,0.f,0.f,0.f,0.f,0.f,0.f,0.f};
  #pragma unroll
  for (int kk = 0; kk < 8; ++kk) {
    const int k = kk * 4 + kb;
    v2f af; af.x = arow[k]; af.y = arow[k + 1];
    acc0 = __builtin_amdgcn_wmma_f32_16x16x4_f32(false, af, false, bfrag[0][kk],
                                                 (short)0, acc0, false, false);
    acc1 = __builtin_amdgcn_wmma_f32_16x16x4_f32(false, af, false, bfrag[1][kk],
                                                 (short)0, acc1, false, false);
  }

  #pragma unroll
  for (int r = 0; r < 8; ++r) {
    const long row = row0 + mo + r;
    if (row < N) {
      Out[row * HD + nc]      = acc0[r];
      Out[row * HD + 16 + nc] = acc1[r];
    }
  }
#else
  for (int r = 0; r < 8; ++r) {
    const long row = row0 + mo + r;
    if (row >= N) continue;
    const float* arow = A + row * HD;
    float a0 = 0.f, a1 = 0.f;
    for (int k = 0; k < HD; ++k) {
      a0 += arow[k] * W[k * HD + nc];
      a1 += arow[k] * W[k * HD + 16 + nc];
    }
    Out[row * HD + nc]      = a0;
    Out[row * HD + 16 + nc] = a1;
  }
#endif
}

// ------------------------------------------------- self-loop term + bias init
__global__ void k_selfinit(const float* __restrict__ hW, const float* __restrict__ nrm,
                           const float* __restrict__ bias, float* __restrict__ agg, int N) {
  long t = (long)blockIdx.x * blockDim.x + threadIdx.x;
  if (t >= (long)N * HD) return;
  long n = t >> 5; int j = (int)(t & (HD - 1));
  float nn = nrm[n];
  agg[t] = hW[t] * nn * nn + bias[j];
}

// ------------------------------------------------- edge scatter: 8 lanes/edge, float4 rows
__global__ void k_edge(const float* __restrict__ hW, const int* __restrict__ src,
                       const int* __restrict__ dst, const float* __restrict__ nrm,
                       float* __restrict__ agg, int E) {
  long t = (long)blockIdx.x * blockDim.x + threadIdx.x;
  long e = t >> 3;
  if (e >= E) return;
  int c = (int)(t & 7);
  int s = src[e], d = dst[e];
  float w = nrm[s] * nrm[d];
  float4 hv = ((const float4*)(hW + (long)s * HD))[c];
  float* out = agg + (long)d * HD + c * 4;
  atomAdd(out + 0, hv.x * w);
  atomAdd(out + 1, hv.y * w);
  atomAdd(out + 2, hv.z * w);
  atomAdd(out + 3, hv.w * w);
}

// ------------------------------------------------- BN stats: per-column sum & sumsq
__global__ void k_bnstats(const float* __restrict__ a, float* __restrict__ sums, int N) {
  __shared__ float s1[8][HD];
  __shared__ float s2[8][HD];
  int j = threadIdx.x & (HD - 1);
  int g = threadIdx.x >> 5;
  float a1 = 0.f, a2 = 0.f;
  for (long row = (long)blockIdx.x * 8 + g; row < N; row += (long)gridDim.x * 8) {
    float v = a[row * HD + j];
    a1 += v; a2 += v * v;
  }
  s1[g][j] = a1; s2[g][j] = a2;
  __syncthreads();
  if (threadIdx.x < HD) {
    float t1 = 0.f, t2 = 0.f;
    #pragma unroll
    for (int k = 0; k < 8; ++k) { t1 += s1[k][j]; t2 += s2[k][j]; }
    atomAdd(&sums[j], t1);
    atomAdd(&sums[HD + j], t2);
  }
}

// ------------------------------------------------- BN apply + ReLU
__global__ void k_bnapply(const float* __restrict__ a, const float* __restrict__ sums,
                          const float* __restrict__ gma, const float* __restrict__ bta,
                          float* __restrict__ out, int N) {
  long t = (long)blockIdx.x * blockDim.x + threadIdx.x;
  if (t >= (long)N * HD) return;
  int j = (int)(t & (HD - 1));
  float invN = 1.0f / (float)N;
  float mean = sums[j] * invN;
  float var  = sums[HD + j] * invN - mean * mean;
  float y = gma[j] * (a[t] - mean) * rsqrtf(var + BN_EPS) + bta[j];
  out[t] = fmaxf(y, 0.0f);
}

// ------------------------------------------------- final FC [N,32]x[32,1]
__global__ void k_fc(const float* __restrict__ h, const float* __restrict__ fcW,
                     const float* __restrict__ fcb, float* __restrict__ out, int N) {
  int n = blockIdx.x * blockDim.x + threadIdx.x;
  if (n >= N) return;
  const float4* hr = (const float4*)(h + (long)n * HD);
  const float4* wr = (const float4*)fcW;
  float acc = 0.f;
  #pragma unroll
  for (int k = 0; k < 8; ++k) {
    float4 a = hr[k], b = wr[k];
    acc += a.x * b.x + a.y * b.y + a.z * b.z + a.w * b.w;
  }
  out[n] = acc + fcb[0];
}

// ----------------------------------------------------------------------------
extern "C" void kernel_launch(void* const* d_in, const int* in_sizes, int n_in,
                              void* d_out, int out_size, void* d_ws, size_t ws_size,
                              hipStream_t stream) {
  const float* x    = (const float*)d_in[0];
  const int*   ei   = (const int*)d_in[1];     // [2,E] flat; edge_attr (d_in[2]) unused
  const float* W1   = (const float*)d_in[3];
  const float* b1   = (const float*)d_in[4];
  const float* g1   = (const float*)d_in[5];
  const float* be1  = (const float*)d_in[6];
  const float* W2   = (const float*)d_in[7];
  const float* b2   = (const float*)d_in[8];
  const float* g2   = (const float*)d_in[9];
  const float* be2  = (const float*)d_in[10];
  const float* W3   = (const float*)d_in[11];
  const float* b3   = (const float*)d_in[12];
  const float* g3   = (const float*)d_in[13];
  const float* be3  = (const float*)d_in[14];
  const float* fcW  = (const float*)d_in[15];
  const float* fcb  = (const float*)d_in[16];

  const int N = in_sizes[0];      // x is [N,1]
  const int E = in_sizes[1] / 2;  // edge_index is [2,E]
  const int* srcI = ei;
  const int* dstI = ei + E;

  // workspace: norm[N] | deg[N] | bufH[N*32] | bufW[N*32] | bufA[N*32] | bnsum[64]
  float* ws    = (float*)d_ws;
  float* nrm   = ws;
  float* deg   = nrm + N;
  float* bufH  = deg + N;
  float* bufW  = bufH + (size_t)N * HD;
  float* bufA  = bufW + (size_t)N * HD;
  float* bnsum = bufA + (size_t)N * HD;

  const int B = 256;
  auto cdiv = [](long a, long b) { return (int)((a + b - 1) / b); };
  const long NH = (long)N * HD;

  // normalization: deg^{-1/2} of A+I
  k_zero  <<<cdiv(N, B), B, 0, stream>>>(deg, N);
  k_degree<<<cdiv(E, B), B, 0, stream>>>(dstI, deg, E);
  k_norm  <<<cdiv(N, B), B, 0, stream>>>(deg, nrm, N);

  const float* Ws[3]  = {W1, W2, W3};
  const float* bs[3]  = {b1, b2, b3};
  const float* gs[3]  = {g1, g2, g3};
  const float* bes[3] = {be1, be2, be3};

  for (int l = 0; l < 3; ++l) {
    if (l == 0)
      k_l1<<<cdiv(NH, B), B, 0, stream>>>(x, W1, bufW, N);
    else
      k_gemm_wmma<<<cdiv(N, 128), B, 0, stream>>>(bufH, Ws[l], bufW, N);

    k_selfinit<<<cdiv(NH, B), B, 0, stream>>>(bufW, nrm, bs[l], bufA, N);
    k_edge    <<<cdiv((long)E * 8, B), B, 0, stream>>>(bufW, srcI, dstI, nrm, bufA, E);

    k_zero   <<<1, 64, 0, stream>>>(bnsum, 2 * HD);
    k_bnstats<<<1024, B, 0, stream>>>(bufA, bnsum, N);
    k_bnapply<<<cdiv(NH, B), B, 0, stream>>>(bufA, bnsum, gs[l], bes[l], bufH, N);
  }

  k_fc<<<cdiv(N, B), B, 0, stream>>>(bufH, fcW, fcb, (float*)d_out, N);
}